// SblNet_86938728006309
// MI455X (gfx1250) — compile-verified
//
#include <hip/hip_runtime.h>
#include <math.h>

typedef __attribute__((ext_vector_type(2))) float v2f;
typedef __attribute__((ext_vector_type(8))) float v8f;

// ---------------------------------------------------------------------------
// Zero workspace (G and c) — required every call since graph replays reuse ws.
// ---------------------------------------------------------------------------
__global__ void zero_kernel(float* __restrict__ p, int n) {
    int i = blockIdx.x * blockDim.x + threadIdx.x;
    if (i < n) p[i] = 0.0f;
}

// ---------------------------------------------------------------------------
// G = X^T X via V_WMMA_F32_16X16X4_F32.
// Grid: (tiles = (P/16)^2, KS split-K chunks). Block: 256 threads = 8 waves.
// Each wave accumulates the full 16x16 fp32 tile over its K-slice (stride 32
// rows, interleaved across the 8 waves); waves reduce through LDS and commit
// with one float atomicAdd per tile element.
//
// fp32 A-matrix 16x4 layout (ISA 7.12.2): lanes 0-15 hold M=lane, v0=K0,v1=K1;
// lanes 16-31 hold v0=K2, v1=K3.  B (4x16) mirrors with N in place of M.
//
// PC is a compile-time P so +P (row 1) and +32*P (next pipeline step /
// prefetch distance) fold into the 24-bit VMEM immediate offsets; the loop
// then carries only two 64-bit pointer adds.  Main loop is guard-free (chunk
// spans are 4-aligned when N%4==0) and software pipelined so loads overlap
// the WMMA issue.
// ---------------------------------------------------------------------------
template <int PC>
__global__ void gram_wmma_kernel(const float* __restrict__ X,
                                 float* __restrict__ G,
                                 int N, int kchunk) {
    const int tilesPerRow = PC >> 4;
    const int ti = blockIdx.x / tilesPerRow;
    const int tj = blockIdx.x % tilesPerRow;
    const int pi = ti << 4;
    const int pj = tj << 4;

    const int lane = threadIdx.x & 31;
    const int wave = threadIdx.x >> 5;
    const int half = lane >> 4;      // 0: rows K0/K1, 1: rows K2/K3
    const int lm   = lane & 15;      // M (for A) / N (for B)

    const int k0 = blockIdx.y * kchunk;                 // k0 multiple of 4
    int ke = k0 + kchunk; if (ke > N) ke = N;
    const int keA = k0 + ((ke - k0) & ~3);              // 4-aligned main bound

    constexpr size_t STRIDE = (size_t)32 * PC;          // 8 waves * 4 K rows
    const float* pA = X + (size_t)(k0 + (wave << 2) + (half << 1)) * PC + pi + lm;
    const float* pB = X + (size_t)(k0 + (wave << 2) + (half << 1)) * PC + pj + lm;

    v8f acc = {};
    int k = k0 + (wave << 2);

    if (k + 3 < keA) {
        v2f a, b;
        a.x = pA[0]; a.y = pA[PC];
        b.x = pB[0]; b.y = pB[PC];
        for (k += 32; k + 3 < keA; k += 32) {
            pA += STRIDE; pB += STRIDE;
            __builtin_prefetch(pA + STRIDE, 0, 1);      // imm-offset prefetch
            __builtin_prefetch(pB + STRIDE, 0, 1);
            v2f an, bn;
            an.x = pA[0]; an.y = pA[PC];
            bn.x = pB[0]; bn.y = pB[PC];
            acc = __builtin_amdgcn_wmma_f32_16x16x4_f32(
                false, a, false, b, (short)0, acc, false, false);
            a = an; b = bn;
        }
        acc = __builtin_amdgcn_wmma_f32_16x16x4_f32(
            false, a, false, b, (short)0, acc, false, false);
        pA += STRIDE; pB += STRIDE;
    }

    // Tail (0-3 leftover rows; at most one wave takes this, never for N%4==0).
    if (k < ke) {
        const int r0 = k + (half << 1);
        const int r1 = r0 + 1;
        v2f a, b;
        a.x = (r0 < ke) ? pA[0]  : 0.0f;
        a.y = (r1 < ke) ? pA[PC] : 0.0f;
        b.x = (r0 < ke) ? pB[0]  : 0.0f;
        b.y = (r1 < ke) ? pB[PC] : 0.0f;
        acc = __builtin_amdgcn_wmma_f32_16x16x4_f32(
            false, a, false, b, (short)0, acc, false, false);
    }

    // C/D layout: VGPR r -> M = r + 8*half (lanes 0-15 / 16-31), N = lm.
    __shared__ float s_part[8][256];
#pragma unroll
    for (int r = 0; r < 8; ++r)
        s_part[wave][(((half << 3) + r) << 4) + lm] = acc[r];
    __syncthreads();

    const int e = threadIdx.x;       // element of the 16x16 tile
    float sum = 0.0f;
#pragma unroll
    for (int w = 0; w < 8; ++w) sum += s_part[w][e];
    atomicAdd(&G[(size_t)(pi + (e >> 4)) * PC + pj + (e & 15)], sum);
}

// Generic runtime-P fallback (same algorithm, addresses not foldable).
__global__ void gram_wmma_kernel_dyn(const float* __restrict__ X,
                                     float* __restrict__ G,
                                     int N, int P, int kchunk) {
    const int tilesPerRow = P >> 4;
    const int ti = blockIdx.x / tilesPerRow;
    const int tj = blockIdx.x % tilesPerRow;
    const int pi = ti << 4;
    const int pj = tj << 4;
    const int lane = threadIdx.x & 31;
    const int wave = threadIdx.x >> 5;
    const int half = lane >> 4;
    const int lm   = lane & 15;

    const int k0 = blockIdx.y * kchunk;
    int ke = k0 + kchunk; if (ke > N) ke = N;
    const int keA = k0 + ((ke - k0) & ~3);

    const size_t stride = (size_t)32 * P;
    const float* pA = X + (size_t)(k0 + (wave << 2) + (half << 1)) * P + pi + lm;
    const float* pB = X + (size_t)(k0 + (wave << 2) + (half << 1)) * P + pj + lm;

    v8f acc = {};
    int k = k0 + (wave << 2);
    for (; k + 3 < keA; k += 32) {
        v2f a, b;
        a.x = pA[0]; a.y = pA[P];
        b.x = pB[0]; b.y = pB[P];
        acc = __builtin_amdgcn_wmma_f32_16x16x4_f32(
            false, a, false, b, (short)0, acc, false, false);
        pA += stride; pB += stride;
    }
    if (k < ke) {
        const int r0 = k + (half << 1);
        const int r1 = r0 + 1;
        v2f a, b;
        a.x = (r0 < ke) ? pA[0] : 0.0f;
        a.y = (r1 < ke) ? pA[P] : 0.0f;
        b.x = (r0 < ke) ? pB[0] : 0.0f;
        b.y = (r1 < ke) ? pB[P] : 0.0f;
        acc = __builtin_amdgcn_wmma_f32_16x16x4_f32(
            false, a, false, b, (short)0, acc, false, false);
    }

    __shared__ float s_part[8][256];
#pragma unroll
    for (int r = 0; r < 8; ++r)
        s_part[wave][(((half << 3) + r) << 4) + lm] = acc[r];
    __syncthreads();
    const int e = threadIdx.x;
    float sum = 0.0f;
#pragma unroll
    for (int w = 0; w < 8; ++w) sum += s_part[w][e];
    atomicAdd(&G[(size_t)(pi + (e >> 4)) * P + pj + (e & 15)], sum);
}

// ---------------------------------------------------------------------------
// c = X^T y.  One block per K-chunk, thread p owns column p (coalesced loads).
// ---------------------------------------------------------------------------
__global__ void xty_kernel(const float* __restrict__ X,
                           const float* __restrict__ y,
                           float* __restrict__ c,
                           int N, int P, int kchunk) {
    const int p = threadIdx.x;
    const int k0 = blockIdx.x * kchunk;
    int ke = k0 + kchunk; if (ke > N) ke = N;
    float s = 0.0f;
    for (int k = k0; k < ke; ++k)
        s += X[(size_t)k * P + p] * y[k];
    atomicAdd(&c[p], s);
}

// ---------------------------------------------------------------------------
// Lambert W0 on [-1/e, 0): identical clamp/seed/Halley as the reference, fp32.
// ---------------------------------------------------------------------------
__device__ __forceinline__ float lambertw_f(float z) {
    const float e = 2.7182818284590452f;
    z = fmaxf(z, -1.0f / e + 1e-6f);
    float w = (z < -0.25f)
                  ? (-1.0f + sqrtf(fmaxf(2.0f * (1.0f + e * z), 0.0f)))
                  : (z * (1.0f - z));
#pragma unroll
    for (int i = 0; i < 20; ++i) {
        float ew  = expf(w);
        float f   = w * ew - z;
        float wp1 = w + 1.0f;
        w = w - f / (ew * wp1 - (w + 2.0f) * f / (2.0f * wp1));
    }
    return w;
}

// ---------------------------------------------------------------------------
// Sequential coordinate descent using G, c.  g_p = dot(x_p, r) maintained by
// rank-1 updates (G row reads are coalesced; G is symmetric so row == column).
// One block of P threads; thread p owns g_p in a register.
// ---------------------------------------------------------------------------
__global__ void cd_kernel(const float* __restrict__ G,
                          const float* __restrict__ c,
                          const float* __restrict__ eta_in,
                          const float* __restrict__ nu,
                          const float* __restrict__ sigma2p,
                          float* __restrict__ eta_out,
                          int P, float tau) {
    const int tid = threadIdx.x;
    __shared__ float s_eta[256];
    __shared__ float s_delta;

    s_eta[tid] = eta_in[tid];
    __syncthreads();

    // g = c - G @ eta   (column read == row read by symmetry; coalesced)
    float g = c[tid];
    for (int q = 0; q < P; ++q)
        g -= G[(size_t)q * P + tid] * s_eta[q];

    float eta_cur = s_eta[tid];
    const float sigma2 = sigma2p[0];

    for (int p = 0; p < P; ++p) {
        if (tid == p) {
            const float x2  = G[(size_t)p * P + p];
            const float ols = g / x2 + eta_cur;   // == dot(x_p, resid_other)/x2
            const float s   = sigma2 / x2;
            const float nup = nu[p];
            const float th  = s * tau / (2.0f * nup);
            const float z   = -(s * tau) / (2.0f * nup * nup)
                              * expf(-fabsf(ols) / nup);
            const float w   = lambertw_f(z);
            const float shrunk = ols + copysignf(nup * w, ols);
            const float en = (fabsf(ols) < th) ? 0.0f : shrunk;
            s_delta  = en - eta_cur;
            eta_cur  = en;
            eta_out[p] = en;
        }
        __syncthreads();
        g -= s_delta * G[(size_t)p * P + tid];    // coalesced row-p read
        __syncthreads();
    }
}

// ---------------------------------------------------------------------------
// Block-wide sum (P power of two); broadcasts identical result to all threads.
// ---------------------------------------------------------------------------
__device__ __forceinline__ float block_reduce_sum(float v, float* red, int P) {
    const int tid = threadIdx.x;
    red[tid] = v;
    __syncthreads();
    for (int off = P >> 1; off > 0; off >>= 1) {
        if (tid < off) red[tid] += red[tid + off];
        __syncthreads();
    }
    float r = red[0];
    __syncthreads();
    return r;
}

// ---------------------------------------------------------------------------
// nu update: 10 Newton iterations on log(nu) with 30-step halving line search.
// cost_p(l) = e^{2l} x2/s2 - (tau/2) exp(-a e^{-l}) - l ;  a = |eta_p|
// g = 2K e^{2l} - T u v - 1 ;  h = 4K e^{2l} - T u v (v-1)  with u=e^{-v},v=a/nu
// The accept test (cont) uses the *summed* cost — scalar, uniform across block.
// ---------------------------------------------------------------------------
__global__ void nu_kernel(const float* __restrict__ G,
                          const float* __restrict__ eta,
                          const float* __restrict__ nu_in,
                          const float* __restrict__ sigma2p,
                          float* __restrict__ nu_out,
                          int P, float tau) {
    __shared__ float red[256];
    const int tid = threadIdx.x;
    const float x2 = G[(size_t)tid * P + tid];
    const float K  = x2 / sigma2p[0];
    const float a  = fabsf(eta[tid]);
    const float T  = 0.5f * tau;

    float l = logf(nu_in[tid]);

    for (int it = 0; it < 10; ++it) {
        float cl = expf(2.0f * l) * K - T * expf(-a * expf(-l)) - l;
        const float oldcost = block_reduce_sum(cl, red, P);
        float step = 1.0f, newcost = INFINITY;

        for (int ls = 0; ls < 30; ++ls) {
            const bool cont = (step >= 1e-8f) &&
                              (newcost > oldcost - 1e-10f * fabsf(oldcost));
            const float v   = a * expf(-l);
            const float u   = expf(-v);
            const float e2l = expf(2.0f * l);
            const float gp  = 2.0f * K * e2l - T * u * v - 1.0f;
            const float hp  = 4.0f * K * e2l - T * u * v * (v - 1.0f);
            const float newl = l - step * gp / hp;
            const float ncl  = expf(2.0f * newl) * K
                               - T * expf(-a * expf(-newl)) - newl;
            const float nc = block_reduce_sum(ncl, red, P);
            if (cont) { l = newl; newcost = nc; step *= 0.5f; }
        }
    }
    nu_out[tid] = expf(l);
}

// ---------------------------------------------------------------------------
extern "C" void kernel_launch(void* const* d_in, const int* in_sizes, int n_in,
                              void* d_out, int out_size, void* d_ws, size_t ws_size,
                              hipStream_t stream) {
    const float* eta = (const float*)d_in[0];
    const float* nu  = (const float*)d_in[1];
    const float* X   = (const float*)d_in[2];
    const float* y   = (const float*)d_in[3];
    const float* s2  = (const float*)d_in[4];

    const int P = in_sizes[0];      // 256
    const int N = in_sizes[3];      // 100000
    const float tau = 1.0f * sqrtf((float)N);

    float* G = (float*)d_ws;                 // P*P floats
    float* c = G + (size_t)P * P;            // P floats
    float* eta_out = (float*)d_out;          // output row 0
    float* nu_out  = eta_out + P;            // output row 1

    const int nz = P * P + P;
    zero_kernel<<<(nz + 255) / 256, 256, 0, stream>>>(G, nz);

    const int tiles = (P / 16) * (P / 16);
    const int KS = 16;
    const int kchunk = (((N + KS - 1) / KS) + 3) & ~3;
    if (P == 256) {
        gram_wmma_kernel<256><<<dim3(tiles, KS), 256, 0, stream>>>(X, G, N, kchunk);
    } else {
        gram_wmma_kernel_dyn<<<dim3(tiles, KS), 256, 0, stream>>>(X, G, N, P, kchunk);
    }

    const int KC = 256;
    const int cchunk = (N + KC - 1) / KC;
    xty_kernel<<<KC, P, 0, stream>>>(X, y, c, N, P, cchunk);

    cd_kernel<<<1, P, 0, stream>>>(G, c, eta, nu, s2, eta_out, P, tau);
    nu_kernel<<<1, P, 0, stream>>>(G, eta_out, nu, s2, nu_out, P, tau);
}